// StackedGRU_64793876628227
// MI455X (gfx1250) — compile-verified
//
#include <hip/hip_runtime.h>
#include <math.h>
#include <stdint.h>

typedef float v2f __attribute__((ext_vector_type(2)));
typedef float v8f __attribute__((ext_vector_type(8)));

#define B_BATCH 64
#define KC 32                  // K-chunk staged in LDS
#define LDS_STRIDE (KC + 4)    // 36 floats = 144B rows: 16B-aligned, bank-spread

// ---- fp32 WMMA: D(16x16) += A(16x4) * B(4x16), full fp32 accumulate ----
__device__ __forceinline__ v8f wmma4(v2f a, v2f b, v8f c) {
    return __builtin_amdgcn_wmma_f32_16x16x4_f32(false, a, false, b, (short)0, c,
                                                 false, false);
}

// Fragment fetch from an LDS tile laid out [rows][LDS_STRIDE] (ISA 7.12.2):
// lanes 0-15 -> row=lane, K={kc,kc+1}; lanes 16-31 -> row=lane-16, K={kc+2,kc+3}.
__device__ __forceinline__ v2f lds_frag(const float* tile, int row0, int kc, int lane) {
    int mi   = lane & 15;
    int half = lane >> 4;
    const float* p = tile + (row0 + mi) * LDS_STRIDE + kc + 2 * half;
    v2f r; r.x = p[0]; r.y = p[1];
    return r;
}

// CDNA5 async copy: 16 bytes global -> LDS, no VGPR bounce, tracked by ASYNCcnt.
__device__ __forceinline__ void async_b128(float* lds_dst, const float* gsrc) {
    uint32_t loff = (uint32_t)(uintptr_t)(__attribute__((address_space(3))) float*)lds_dst;
    asm volatile("global_load_async_to_lds_b128 %0, %1, off"
                 :: "v"(loff), "v"(gsrc)
                 : "memory");
}
__device__ __forceinline__ void wait_async0() {
    asm volatile("s_wait_asynccnt 0x0" ::: "memory");
}

// ============================================================================
// Kernel 1: input projections for a whole layer (parallel over all T*B rows).
//   Gx[(row*3+g)*H + n] = A[row,:] . W_g[n,:] + b_g[n],  row = t*B + b
// Block tile 64x64, one gate per blockIdx.z; 8 waves = 4(M)x2(N), each wave a
// 16x32 slab. Double-buffered async LDS staging, one barrier per K-chunk.
// ============================================================================
__global__ __launch_bounds__(256)
void gru_in_proj(const float* __restrict__ A, long stride_t, long stride_b,
                 const float* __restrict__ Wr, const float* __restrict__ Wz,
                 const float* __restrict__ Wn,
                 const float* __restrict__ br, const float* __restrict__ bz,
                 const float* __restrict__ bn,
                 float* __restrict__ Gx, int K, int H)
{
    __shared__ __align__(16) float lA[2][64 * LDS_STRIDE];
    __shared__ __align__(16) float lB[2][64 * LDS_STRIDE];

    const int g = blockIdx.z;
    const float* W    = (g == 0) ? Wr : (g == 1) ? Wz : Wn;
    const float* bias = (g == 0) ? br : (g == 1) ? bz : bn;

    const int rowBase = blockIdx.x * 64;
    const int nBase   = blockIdx.y * 64;
    const int tid  = threadIdx.x;
    const int lane = tid & 31;
    const int wave = tid >> 5;
    const int wm = wave & 3;    // M tile 0..3
    const int wn = wave >> 2;   // N half 0..1 (32 cols each)

    v8f acc0 = {0.f,0.f,0.f,0.f,0.f,0.f,0.f,0.f};
    v8f acc1 = acc0;

    const int sr = tid >> 3;          // 0..31 staging row
    const int kq = (tid & 7) * 4;     // float4 column
    const int NC = K / KC;

    auto issue = [&](int cc, int buf) {
        const int k0 = cc * KC;
        #pragma unroll
        for (int p = 0; p < 2; ++p) {
            const int row = sr + p * 32;
            const long R  = rowBase + row;
            async_b128(&lA[buf][row * LDS_STRIDE + kq],
                       A + (R / B_BATCH) * stride_t + (R % B_BATCH) * stride_b + k0 + kq);
            async_b128(&lB[buf][row * LDS_STRIDE + kq],
                       W + (long)(nBase + row) * K + k0 + kq);
        }
    };

    issue(0, 0);
    for (int c = 0; c < NC; ++c) {
        const int buf = c & 1;
        wait_async0();          // my copies for chunk c are in LDS
        __syncthreads();        // everyone's copies landed; buf^1 free to refill
        if (c + 1 < NC) issue(c + 1, buf ^ 1);
        const float* tA = lA[buf];
        const float* tB = lB[buf];
        #pragma unroll
        for (int k4 = 0; k4 < KC / 4; ++k4) {
            const int kc = k4 * 4;
            v2f a  = lds_frag(tA, wm * 16, kc, lane);
            v2f b0 = lds_frag(tB, wn * 32,      kc, lane);
            v2f b1 = lds_frag(tB, wn * 32 + 16, kc, lane);
            acc0 = wmma4(a, b0, acc0);
            acc1 = wmma4(a, b1, acc1);
        }
    }

    const int half = lane >> 4;
    const int ni   = lane & 15;
    #pragma unroll
    for (int r = 0; r < 8; ++r) {
        const int  m   = wm * 16 + r + half * 8;
        const long row = rowBase + m;
        const int  n0  = nBase + wn * 32 + ni;
        const int  n1  = n0 + 16;
        float* dst = Gx + (row * 3 + g) * (long)H;
        dst[n0] = acc0[r] + bias[n0];
        dst[n1] = acc1[r] + bias[n1];
    }
}

// ============================================================================
// Kernel 2: one recurrent timestep (serial critical path).
// Block tile: all 64 batch rows x 32 hidden units, 3 gates fused; 8 waves =
// 4(M)x2(N), each wave 3 accumulators sharing one A fragment -> 3 v_wmma per
// K-step. Async double-buffered staging hides L2->LDS latency of W_h behind
// the WMMA stream. Epilogue fuses sigmoid/tanh gate math.
// ============================================================================
__global__ __launch_bounds__(256)
void gru_step(const float* __restrict__ hprev,
              const float* __restrict__ Whr, const float* __restrict__ Whz,
              const float* __restrict__ Whn,
              const float* __restrict__ bhr, const float* __restrict__ bhz,
              const float* __restrict__ bhn,
              const float* __restrict__ Gxt,   // Gx + t*B*3*H
              float* __restrict__ hnext,
              float* __restrict__ ybase, long y_mstride,
              int H)
{
    __shared__ __align__(16) float lA [2][64 * LDS_STRIDE];
    __shared__ __align__(16) float lBr[2][32 * LDS_STRIDE];
    __shared__ __align__(16) float lBz[2][32 * LDS_STRIDE];
    __shared__ __align__(16) float lBn[2][32 * LDS_STRIDE];

    const int nBase = blockIdx.x * 32;
    const int tid  = threadIdx.x;
    const int lane = tid & 31;
    const int wave = tid >> 5;
    const int wm = wave & 3;
    const int wn = wave >> 2;   // 0..1 -> 16 cols each

    v8f ar = {0.f,0.f,0.f,0.f,0.f,0.f,0.f,0.f};
    v8f az = ar, an = ar;

    const int sr = tid >> 3;
    const int kq = (tid & 7) * 4;
    const int K  = H;
    const int NC = K / KC;

    auto issue = [&](int cc, int buf) {
        const int k0 = cc * KC;
        #pragma unroll
        for (int p = 0; p < 2; ++p) {
            const int row = sr + p * 32;
            async_b128(&lA[buf][row * LDS_STRIDE + kq], hprev + (long)row * H + k0 + kq);
        }
        const long wro = (long)(nBase + sr) * H + k0 + kq;
        async_b128(&lBr[buf][sr * LDS_STRIDE + kq], Whr + wro);
        async_b128(&lBz[buf][sr * LDS_STRIDE + kq], Whz + wro);
        async_b128(&lBn[buf][sr * LDS_STRIDE + kq], Whn + wro);
    };

    issue(0, 0);
    for (int c = 0; c < NC; ++c) {
        const int buf = c & 1;
        wait_async0();
        __syncthreads();
        if (c + 1 < NC) issue(c + 1, buf ^ 1);
        const float* tA = lA[buf];
        const float* tR = lBr[buf];
        const float* tZ = lBz[buf];
        const float* tN = lBn[buf];
        #pragma unroll
        for (int k4 = 0; k4 < KC / 4; ++k4) {
            const int kc = k4 * 4;
            v2f a  = lds_frag(tA, wm * 16, kc, lane);
            v2f br = lds_frag(tR, wn * 16, kc, lane);
            v2f bz = lds_frag(tZ, wn * 16, kc, lane);
            v2f bn = lds_frag(tN, wn * 16, kc, lane);
            ar = wmma4(a, br, ar);
            az = wmma4(a, bz, az);
            an = wmma4(a, bn, an);
        }
    }

    const int half = lane >> 4;
    const int n    = nBase + wn * 16 + (lane & 15);
    const float vbr = bhr[n], vbz = bhz[n], vbn = bhn[n];

    #pragma unroll
    for (int r8 = 0; r8 < 8; ++r8) {
        const int m = wm * 16 + r8 + half * 8;
        const float* gx = Gxt + (long)m * 3 * H;
        const float gxr = gx[0 * H + n];
        const float gxz = gx[1 * H + n];
        const float gxn = gx[2 * H + n];
        const float hv  = hprev[(long)m * H + n];
        const float rg = 1.f / (1.f + __expf(-(gxr + ar[r8] + vbr)));
        const float zg = 1.f / (1.f + __expf(-(gxz + az[r8] + vbz)));
        const float ng = tanhf(gxn + rg * (an[r8] + vbn));
        const float hv2 = (1.f - zg) * ng + zg * hv;
        hnext[(long)m * H + n] = hv2;
        ybase[(long)m * y_mstride + n] = hv2;
    }
}

// ============================================================================
extern "C" void kernel_launch(void* const* d_in, const int* in_sizes, int n_in,
                              void* d_out, int out_size, void* d_ws, size_t ws_size,
                              hipStream_t stream)
{
    (void)in_sizes; (void)n_in; (void)out_size; (void)ws_size;
    const int T = 256, Bn = 64, IN = 512, H = 1024, L = 2;

    const float* Xt = (const float*)d_in[0];
    float* out = (float*)d_out;

    // workspace layout (floats)
    float* gx = (float*)d_ws;                        // [T,B,3,H]   201 MB
    float* x1 = gx + (size_t)T * Bn * 3 * H;         // [T,B,H]      67 MB
    float* hA = x1 + (size_t)T * Bn * H;             // [B,H]
    float* hB = hA + (size_t)Bn * H;                 // [B,H]

    for (int l = 0; l < L; ++l) {
        const float* const* P = (const float* const*)(d_in + 1 + l * 12);
        const float *W_ir = P[0],  *b_ir = P[1],  *W_hr = P[2],  *b_hr = P[3];
        const float *W_iz = P[4],  *b_iz = P[5],  *W_hz = P[6],  *b_hz = P[7];
        const float *W_in = P[8],  *b_in = P[9],  *W_hn = P[10], *b_hn = P[11];

        const int   K        = (l == 0) ? IN : H;
        const float* A       = (l == 0) ? Xt : x1;
        const long  stride_t = (l == 0) ? (long)IN      : (long)Bn * H;
        const long  stride_b = (l == 0) ? (long)T * IN  : (long)H;

        // 1) all input projections for this layer (parallel over T)
        dim3 g1((T * Bn) / 64, H / 64, 3);
        gru_in_proj<<<g1, 256, 0, stream>>>(A, stride_t, stride_b,
                                            W_ir, W_iz, W_in, b_ir, b_iz, b_in,
                                            gx, K, H);

        // 2) serial recurrence
        hipMemsetAsync(hA, 0, (size_t)Bn * H * sizeof(float), stream);
        for (int t = 0; t < T; ++t) {
            const float* hin  = (t & 1) ? hB : hA;
            float*       hout = (t & 1) ? hA : hB;
            float* ybase; long ms;
            if (l == 0) { ybase = x1 + (size_t)t * Bn * H; ms = H; }
            else        { ybase = out + (size_t)t * H;     ms = (long)T * H; }
            gru_step<<<dim3(H / 32), 256, 0, stream>>>(hin,
                    W_hr, W_hz, W_hn, b_hr, b_hz, b_hn,
                    gx + (size_t)t * Bn * 3 * H,
                    hout, ybase, ms, H);
        }
        // t = 255 (odd) wrote hA -> final hidden state of this layer
        hipMemcpyAsync(out + (size_t)Bn * T * H + (size_t)l * Bn * H, hA,
                       (size_t)Bn * H * sizeof(float),
                       hipMemcpyDeviceToDevice, stream);
    }
}